// ViewSigLayer_52544629899939
// MI455X (gfx1250) — compile-verified
//
#include <hip/hip_runtime.h>

// Depth-4 path signature (B=1024, L=256, C=12) for MI455X / gfx1250.
//
// Chen's relation re-factored so every level-4 contribution is
// prefix(t) (x) dx(t):   S4 = sum_t A(t) (x) dx(t),
//   A(t)[c1c2c3] = S3prev + S2prev[c1c2]*dx[c3]/2
//                + S1prev[c1]*dx[c2]*dx[c3]/6 + dx[c1]dx[c2]dx[c3]/24
// i.e. a 1728 x T @ T x 12 matmul (T = 255) per batch element, run on
// V_WMMA_F32_16X16X4_F32 (fp32 kept: signature terms accumulate over 255
// steps; half-precision inputs would wreck the low levels).
// S1/S2/S3 prefixes are cheap sequential VALU work in LDS.
//
// 128 threads = 4 waves; 108 M-tiles split 27/wave exactly -> every WMMA is
// unconditional (no saveexec around the matrix ops).  sdx is padded to 16
// columns so B-fragments load unconditionally too.

typedef __attribute__((ext_vector_type(2))) float v2f;
typedef __attribute__((ext_vector_type(8))) float v8f;

#define CCH    12        // channels
#define LEN    256       // path length
#define NT     255       // number of increments (K of the big matmul)
#define C2     144       // 12^2
#define C3     1728      // 12^3
#define OUTB   22620     // 12+144+1728+20736 per batch
#define TC     8         // time-chunk (K tile held in LDS)
#define NCHUNK 32        // 32*8 = 256 padded steps (t >= 255 padded with dx=0)
#define ASTR   1744      // LDS row stride for A-stream (1728+16, bank pad)
#define DXW    16        // sdx padded row width (cols 12..15 always zero)
#define NTHR   128       // threads per block (4 waves)
#define NWAVE  4
#define TPW    27        // M-tiles per wave: 108/4, exact
#define EPT    14        // ceil(1728/128) elements per thread (sequential phase)

__global__ __launch_bounds__(NTHR)
void sig_depth4_kernel(const float* __restrict__ x, float* __restrict__ out)
{
    __shared__ float sA[TC * ASTR];   // A(t) stream for current chunk (Kc x M)
    __shared__ float sS3[C3];         // running level-3 signature
    __shared__ float sS2[2][C2];      // double-buffered level-2 prefix
    __shared__ float sS1[2][CCH];     // double-buffered level-1 prefix
    __shared__ float sdx[TC * DXW];   // increments, zero-padded to 16 cols

    const int tid  = threadIdx.x;
    const int lane = tid & 31;
    const int wave = tid >> 5;
    const int b    = blockIdx.x;
    const float* xb = x + (size_t)b * (CCH * LEN);

    // ---- init shared state --------------------------------------------------
    for (int i = tid; i < C3; i += NTHR) sS3[i] = 0.f;
    for (int u = tid; u < C2; u += NTHR) { sS2[0][u] = 0.f; sS2[1][u] = 0.f; }
    if (tid < CCH) { sS1[0][tid] = 0.f; sS1[1][tid] = 0.f; }

    v8f acc[TPW];
    #pragma unroll
    for (int j = 0; j < TPW; ++j) { v8f z = {}; acc[j] = z; }

    // Packed (c1,c2,c3) per element handled in the sequential phase.
    unsigned pidx[EPT];
    #pragma unroll
    for (int k = 0; k < EPT; ++k) {
        int i = tid + (k << 7);
        if (i < C3) {
            unsigned c1 = (unsigned)(i / C2);
            unsigned c2 = (unsigned)((i / CCH) % CCH);
            unsigned c3 = (unsigned)(i % CCH);
            pidx[k] = c1 | (c2 << 8) | (c3 << 16);
        } else {
            pidx[k] = 0u;
        }
    }

    __syncthreads();

    const float k2 = 0.5f, k6 = 1.f / 6.f, k24 = 1.f / 24.f;

    for (int ch = 0; ch < NCHUNK; ++ch) {
        // ---- stage increments dx(t) (128 threads cover 8x16 exactly) --------
        {
            int t = ch * TC + (tid >> 4);
            int c = tid & 15;
            float d = 0.f;
            if (t < NT && c < CCH) d = xb[c * LEN + t + 1] - xb[c * LEN + t];
            sdx[tid] = d;
        }
        __syncthreads();

        // ---- sequential prefix phase: build A(t) stream, advance S1/S2/S3 ---
        for (int tt = 0; tt < TC; ++tt) {
            const float* dxp = &sdx[tt * DXW];
            const float* S1o = sS1[tt & 1];       // ch*TC even -> parity = tt&1
            const float* S2o = sS2[tt & 1];
            float*       S1n = sS1[(tt + 1) & 1];
            float*       S2n = sS2[(tt + 1) & 1];

            #pragma unroll
            for (int k = 0; k < EPT; ++k) {
                int i = tid + (k << 7);
                if (i < C3) {
                    unsigned pk = pidx[k];
                    int c1 = (int)(pk & 0xffu);
                    int c2 = (int)((pk >> 8) & 0xffu);
                    int c3 = (int)(pk >> 16);
                    float p  = dxp[c1];
                    float q  = dxp[c2];
                    float r  = dxp[c3];
                    float s1 = S1o[c1];
                    float s2 = S2o[c1 * CCH + c2];
                    float s3 = sS3[i];
                    float qr = q * r;
                    // A(t): prefix of every level-4 Chen term (dx factored right)
                    sA[tt * ASTR + i] = s3 + k2 * s2 * r + k6 * s1 * qr + k24 * p * qr;
                    // Chen: S3 += (S2 + S1(x)dx/2 + dx^2/6) (x) dx
                    sS3[i] = s3 + (s2 + k2 * s1 * q + k6 * p * q) * r;
                }
            }
            #pragma unroll
            for (int u = tid; u < C2; u += NTHR) {
                int c1 = u / CCH, c2c = u % CCH;
                S2n[u] = S2o[u] + (S1o[c1] + k2 * dxp[c1]) * dxp[c2c];
            }
            if (tid < CCH) S1n[tid] = S1o[tid] + dxp[tid];
            __syncthreads();
        }

        // ---- WMMA phase: acc(1728x12) += A(1728 x TC) * dx(TC x 12) ---------
        // fp32 WMMA 16x16x4.  A: lanes 0-15 -> M=lane, K=0/1 in v0/v1;
        // lanes 16-31 -> M=lane-16, K=2/3.  B mirrored (K per VGPR/half-wave,
        // N = lane&15; cols 12..15 are zero by construction of sdx).
        {
            const int col   = lane & 15;
            const int khalf = (lane >= 16) ? 2 : 0;
            const int mbase = wave * (TPW * 16);   // contiguous 432 rows/wave
            #pragma unroll
            for (int kb = 0; kb < TC / 4; ++kb) {
                int t0 = kb * 4 + khalf;
                v2f bf;
                bf.x = sdx[t0 * DXW + col];
                bf.y = sdx[(t0 + 1) * DXW + col];
                #pragma unroll
                for (int j = 0; j < TPW; ++j) {
                    int row = mbase + j * 16 + col;
                    v2f af;
                    af.x = sA[t0 * ASTR + row];
                    af.y = sA[(t0 + 1) * ASTR + row];
                    acc[j] = __builtin_amdgcn_wmma_f32_16x16x4_f32(
                        false, af, false, bf, (short)0, acc[j], false, false);
                }
            }
        }
        __syncthreads();
    }

    // ---- writeback ---------------------------------------------------------
    float* ob = out + (size_t)b * OUTB;

    // After 256 padded steps the final S1/S2 live in parity 0.
    if (tid < CCH) ob[tid] = sS1[0][tid];
    for (int u = tid; u < C2; u += NTHR) ob[CCH + u] = sS2[0][u];
    for (int i = tid; i < C3; i += NTHR) ob[CCH + C2 + i] = sS3[i];

    // S4 from accumulators. C/D layout: VGPR v -> M = v (+8 for lanes 16-31),
    // N = lane&15; only N<12 is a real column.
    const int col   = lane & 15;
    const int rhalf = (lane >= 16) ? 8 : 0;
    if (col < CCH) {
        #pragma unroll
        for (int j = 0; j < TPW; ++j) {
            int rbase = wave * (TPW * 16) + j * 16 + rhalf;
            #pragma unroll
            for (int v = 0; v < 8; ++v) {
                int row = rbase + v;
                ob[CCH + C2 + C3 + row * CCH + col] = acc[j][v];
            }
        }
    }
}

extern "C" void kernel_launch(void* const* d_in, const int* in_sizes, int n_in,
                              void* d_out, int out_size, void* d_ws, size_t ws_size,
                              hipStream_t stream)
{
    const float* x = (const float*)d_in[0];
    float* out = (float*)d_out;
    const int B = in_sizes[0] / (CCH * LEN);   // 1024
    hipLaunchKernelGGL(sig_depth4_kernel, dim3(B), dim3(NTHR), 0, stream, x, out);
}